// RelationalGCN_39805756900151
// MI455X (gfx1250) — compile-verified
//
#include <hip/hip_runtime.h>

// ---------------------------------------------------------------------------
// RGCN (basis decomposition) for MI455X / gfx1250, wave32, WMMA f32 16x16x4.
// ---------------------------------------------------------------------------

typedef float v2f __attribute__((ext_vector_type(2)));
typedef float v8f __attribute__((ext_vector_type(8)));

#define N_NODES   8192
#define N_EDGES   16384
#define DMODEL    128
#define NREL      64
#define NBASIS    16

// Worst-case padded edge count: each relation rounded up to multiple of 16.
#define MAX_TILES 1088                    // >= ceil((16384 + 64*15)/16)
#define SORT_CAP  (MAX_TILES * 16)        // 17408

// Workspace layout (in 4-byte words):
//   [0)                 counts   : N_NODES*NREL ints        (524288)
//   [+524288)           hist     : NREL ints                (64)
//   [+524352)           cursor   : NREL ints                (64)
//   [+524416)           sorted   : SORT_CAP ints            (17408)
//   [+541824)           tile_rel : MAX_TILES ints           (1088)
//   [542976, aligned)   fw_t     : NREL*DMODEL*DMODEL floats (4MB), fw_t[r][n][k]
#define WS_COUNTS_OFF   0
#define WS_HIST_OFF     (N_NODES * NREL)
#define WS_CURSOR_OFF   (WS_HIST_OFF + NREL)
#define WS_SORTED_OFF   (WS_CURSOR_OFF + NREL)
#define WS_TILEREL_OFF  (WS_SORTED_OFF + SORT_CAP)
#define WS_FWT_OFF      542976            // 256-word aligned, >= WS_TILEREL_OFF+MAX_TILES

// ---------------------------------------------------------------------------
__global__ void rgcn_init_zero(int* __restrict__ p, int n) {
    int i = blockIdx.x * blockDim.x + threadIdx.x;
    if (i < n) p[i] = 0;
}

__global__ void rgcn_init_neg1(int* __restrict__ p, int n) {
    int i = blockIdx.x * blockDim.x + threadIdx.x;
    if (i < n) p[i] = -1;
}

// fw_t[r][n][k] = sum_b w_comp[r][b] * weight[b][k][n]   (transposed full_weight)
__global__ void rgcn_fwt(const float* __restrict__ w_comp,
                         const float* __restrict__ weight,
                         float* __restrict__ fw_t) {
    int idx = blockIdx.x * blockDim.x + threadIdx.x;   // 0 .. 64*128*128-1
    int k = idx & 127;
    int n = (idx >> 7) & 127;
    int r = idx >> 14;
    float acc = 0.0f;
#pragma unroll
    for (int b = 0; b < NBASIS; ++b)
        acc = fmaf(w_comp[r * NBASIS + b], weight[b * (DMODEL * DMODEL) + k * DMODEL + n], acc);
    fw_t[idx] = acc;
}

// histogram over relations + per-(tgt,rel) neighbor counts
__global__ void rgcn_hist(const int* __restrict__ deprel,
                          const int* __restrict__ edge_index,
                          int* __restrict__ hist,
                          int* __restrict__ counts) {
    int e = blockIdx.x * blockDim.x + threadIdx.x;
    if (e < N_EDGES) {
        int r = deprel[e];
        int t = edge_index[N_EDGES + e];
        atomicAdd(&hist[r], 1);
        atomicAdd(&counts[t * NREL + r], 1);
    }
}

// serial 64-entry scan with padding to 16; also fills tile -> relation map
__global__ void rgcn_scan(const int* __restrict__ hist,
                          int* __restrict__ cursor,
                          int* __restrict__ tile_rel) {
    if (threadIdx.x == 0 && blockIdx.x == 0) {
        int run = 0;
        for (int r = 0; r < NREL; ++r) {
            cursor[r] = run;
            int nt = (hist[r] + 15) >> 4;
            int t0 = run >> 4;
            for (int t = 0; t < nt; ++t) tile_rel[t0 + t] = r;
            run += nt << 4;
        }
    }
}

// bucket-scatter edge ids into relation-sorted order
__global__ void rgcn_scatter(const int* __restrict__ deprel,
                             int* __restrict__ cursor,
                             int* __restrict__ sorted) {
    int e = blockIdx.x * blockDim.x + threadIdx.x;
    if (e < N_EDGES) {
        int r = deprel[e];
        int pos = atomicAdd(&cursor[r], 1);
        sorted[pos] = e;
    }
}

// ---------------------------------------------------------------------------
// out[row][n] = sum_k inp[row][k] * self_weight[n][k] + bias[n]
// One block = 16 rows x 128 cols; 8 waves, each owns a 16-col N-tile.
// ---------------------------------------------------------------------------
__global__ __launch_bounds__(256)
void rgcn_self_gemm(const float* __restrict__ inp,
                    const float* __restrict__ self_w,
                    const float* __restrict__ bias,
                    float* __restrict__ out) {
    __shared__ float X[16][DMODEL];        // 8 KB A-tile
    const int tid  = threadIdx.x;
    const int row0 = blockIdx.x * 16;

    {   // cooperative coalesced load: 16 threads per row, 8 floats each
        int m = tid >> 4;
        int i = (tid & 15) << 3;
        const float4* sp = (const float4*)&inp[(size_t)(row0 + m) * DMODEL + i];
        float4 a0 = sp[0], a1 = sp[1];
        *(float4*)&X[m][i]     = a0;
        *(float4*)&X[m][i + 4] = a1;
    }
    __syncthreads();

    const int wave = tid >> 5;             // 0..7 -> N-tile
    const int lane = tid & 31;
    const int half = lane >> 4;            // 0: K = k0,k0+1 ; 1: K = k0+2,k0+3
    const int l16  = lane & 15;
    const int n    = wave * 16 + l16;      // global column

    const v2f* xp = (const v2f*)&X[l16][0];
    const float* bw = &self_w[(size_t)n * DMODEL];   // B[k][n] = self_w[n][k]

    v8f c = {};
#pragma unroll
    for (int k0 = 0; k0 < DMODEL; k0 += 4) {
        int kb = k0 + half * 2;
        v2f a  = xp[kb >> 1];                         // A: X[M=l16][kb..kb+1]
        v2f b2 = *(const v2f*)(bw + kb);              // contiguous, 8B aligned
        c = __builtin_amdgcn_wmma_f32_16x16x4_f32(false, a, false, b2,
                                                  (short)0, c, false, false);
    }

    const float bn = bias[n];
#pragma unroll
    for (int i = 0; i < 8; ++i) {
        int M = i + half * 8;
        out[(size_t)(row0 + M) * DMODEL + n] = c[i] + bn;
    }
}

// ---------------------------------------------------------------------------
// One block per 16-edge tile (all same relation r). A = 1/count-scaled gathered
// rows of inp; B = fw_t[r] (pre-transposed); result atomically added to out[tgt].
// ---------------------------------------------------------------------------
__global__ __launch_bounds__(256)
void rgcn_edge_gemm(const float* __restrict__ inp,
                    const float* __restrict__ fw_t,
                    const int* __restrict__ edge_index,
                    const int* __restrict__ counts,
                    const int* __restrict__ sorted,
                    const int* __restrict__ tile_rel,
                    float* __restrict__ out) {
    const int t = blockIdx.x;
    const int r = tile_rel[t];
    if (r < 0) return;                       // unused padding tile

    __shared__ float X[16][DMODEL];          // scaled A-tile
    __shared__ int   Tgt[16];

    const int tid = threadIdx.x;
    {   // gather + scale: 16 threads per edge-row, 8 floats each
        int m = tid >> 4;
        int i = (tid & 15) << 3;
        int e = sorted[t * 16 + m];
        int src = 0, tgt = -1;
        float scale = 0.0f;
        if (e >= 0) {
            src = edge_index[e];
            tgt = edge_index[N_EDGES + e];
            scale = 1.0f / (float)counts[tgt * NREL + r];
        }
        if ((tid & 15) == 0) Tgt[m] = tgt;
        const float4* sp = (const float4*)&inp[(size_t)src * DMODEL + i];
        float4 a0 = sp[0], a1 = sp[1];       // src==0 default keeps load safe
        a0.x *= scale; a0.y *= scale; a0.z *= scale; a0.w *= scale;
        a1.x *= scale; a1.y *= scale; a1.z *= scale; a1.w *= scale;
        *(float4*)&X[m][i]     = a0;
        *(float4*)&X[m][i + 4] = a1;
    }
    __syncthreads();

    const int wave = tid >> 5;
    const int lane = tid & 31;
    const int half = lane >> 4;
    const int l16  = lane & 15;
    const int n    = wave * 16 + l16;

    const v2f*   xp = (const v2f*)&X[l16][0];
    const float* bw = &fw_t[((size_t)r * DMODEL + n) * DMODEL];  // fw_t[r][n][k]

    v8f c = {};
#pragma unroll
    for (int k0 = 0; k0 < DMODEL; k0 += 4) {
        int kb = k0 + half * 2;
        v2f a  = xp[kb >> 1];
        v2f b2 = *(const v2f*)(bw + kb);
        c = __builtin_amdgcn_wmma_f32_16x16x4_f32(false, a, false, b2,
                                                  (short)0, c, false, false);
    }

#pragma unroll
    for (int i = 0; i < 8; ++i) {
        int M  = i + half * 8;
        int tg = Tgt[M];
        if (tg >= 0) atomicAdd(&out[(size_t)tg * DMODEL + n], c[i]);
    }
}

// ---------------------------------------------------------------------------
extern "C" void kernel_launch(void* const* d_in, const int* in_sizes, int n_in,
                              void* d_out, int out_size, void* d_ws, size_t ws_size,
                              hipStream_t stream) {
    const float* inp        = (const float*)d_in[0];
    const int*   deprel     = (const int*)  d_in[1];
    const int*   edge_index = (const int*)  d_in[2];
    const float* weight     = (const float*)d_in[3];
    const float* w_comp     = (const float*)d_in[4];
    const float* self_w     = (const float*)d_in[5];
    const float* bias       = (const float*)d_in[6];
    float*       out        = (float*)d_out;

    int*   wsw      = (int*)d_ws;
    int*   counts   = wsw + WS_COUNTS_OFF;
    int*   hist     = wsw + WS_HIST_OFF;
    int*   cursor   = wsw + WS_CURSOR_OFF;
    int*   sorted   = wsw + WS_SORTED_OFF;
    int*   tile_rel = wsw + WS_TILEREL_OFF;
    float* fw_t     = (float*)wsw + WS_FWT_OFF;

    // 1. reset per-call scratch (deterministic across graph replays)
    {
        int nz = N_NODES * NREL + NREL;                 // counts + hist (contiguous)
        rgcn_init_zero<<<(nz + 255) / 256, 256, 0, stream>>>(counts, nz);
        int nn = SORT_CAP + MAX_TILES;                  // sorted + tile_rel (contiguous)
        rgcn_init_neg1<<<(nn + 255) / 256, 256, 0, stream>>>(sorted, nn);
    }
    // 2. transposed basis-combined weights  fw_t[r][n][k]
    rgcn_fwt<<<(NREL * DMODEL * DMODEL) / 256, 256, 0, stream>>>(w_comp, weight, fw_t);
    // 3. relation histogram + (tgt,rel) counts
    rgcn_hist<<<N_EDGES / 256, 256, 0, stream>>>(deprel, edge_index, hist, counts);
    // 4. padded prefix-sum + tile->relation map
    rgcn_scan<<<1, 32, 0, stream>>>(hist, cursor, tile_rel);
    // 5. bucket edges by relation
    rgcn_scatter<<<N_EDGES / 256, 256, 0, stream>>>(deprel, cursor, sorted);
    // 6. self-loop GEMM (initializes out, bias folded in)
    rgcn_self_gemm<<<N_NODES / 16, 256, 0, stream>>>(inp, self_w, bias, out);
    // 7. relation-tiled edge GEMM, atomic scatter-accumulate
    rgcn_edge_gemm<<<MAX_TILES, 256, 0, stream>>>(inp, fw_t, edge_index, counts,
                                                  sorted, tile_rel, out);
}